// EINModel_v3_78374563217906
// MI455X (gfx1250) — compile-verified
//
#include <hip/hip_runtime.h>
#include <hip/hip_bf16.h>
#include <math.h>

#define N_NODES   20000
#define N_EDGES   320000
#define N_GRAPHS  64
#define IN_DIM    128
#define DIM_H     32
#define HEADS     4
#define EDGE_DIM  16
#define HC        128            // HEADS*DIM_H
#define FINAL_DIM 10

typedef __attribute__((ext_vector_type(2))) float v2f;
typedef __attribute__((ext_vector_type(8))) float v8f;
typedef int v4i_ __attribute__((vector_size(16)));   // matches builtin's V4i param

#if __has_builtin(__builtin_amdgcn_global_load_async_to_lds_b128)
#define HAVE_ASYNC_LDS 1
#endif

// ---------- order-monotonic float<->uint key for native atomicMax ----------
__device__ __forceinline__ unsigned fkey(float f) {
    unsigned b = __float_as_uint(f);
    return (b & 0x80000000u) ? ~b : (b | 0x80000000u);
}
__device__ __forceinline__ float funkey(unsigned k) {
    unsigned b = (k & 0x80000000u) ? (k & 0x7FFFFFFFu) : ~k;
    return __uint_as_float(b);
}

// =====================================================================
// Fused Q/K/V projection: fp32 WMMA 16x16x4. One block = 16-row strip of A.
// 8 waves cover 24 output tiles (3 matrices x 8 N-tiles of 16).
// A strip staged to LDS via async global->LDS b128 (ASYNCcnt) when the
// toolchain exposes the builtin, else vectorized float4 + ds_store_b128.
// =====================================================================
__global__ __launch_bounds__(256) void qkv_wmma(
    const float* __restrict__ A,                       // [M,128] row-major
    const float* __restrict__ Bq, const float* __restrict__ Bk,
    const float* __restrict__ Bv,                      // [128,128]
    float* __restrict__ Q, float* __restrict__ K, float* __restrict__ V)
{
    __shared__ float As[16 * 132];   // padded stride (528B = 33*16B): conflict-free, 16B-aligned rows
    const int m0  = blockIdx.x * 16;
    const int tid = threadIdx.x;

    // 16 rows x 128 floats = 512 x 16B chunks; 2 chunks per thread.
#ifdef HAVE_ASYNC_LDS
    for (int i = tid; i < 512; i += 256) {
        const int r = i >> 5, c4 = i & 31;
        v4i_* gp = (v4i_*)(A + (size_t)(m0 + r) * 128 + c4 * 4);
        v4i_* lp = (v4i_*)(&As[r * 132 + c4 * 4]);
        __builtin_amdgcn_global_load_async_to_lds_b128(
            (__attribute__((address_space(1))) v4i_*)gp,
            (__attribute__((address_space(3))) v4i_*)lp,
            0, 0);
    }
#if __has_builtin(__builtin_amdgcn_s_wait_asynccnt)
    __builtin_amdgcn_s_wait_asynccnt(0);
#else
    asm volatile("s_wait_asynccnt 0x0" ::: "memory");
#endif
#else
    for (int i = tid; i < 512; i += 256) {
        const int r = i >> 5, c4 = i & 31;
        const float4 val = ((const float4*)A)[(size_t)(m0 + r) * 32 + c4];
        *(float4*)&As[r * 132 + c4 * 4] = val;
    }
#endif
    __syncthreads();

    const int wave = tid >> 5;
    const int lane = tid & 31;
    const int half = lane >> 4;       // 0 or 1
    const int l16  = lane & 15;

    for (int t = wave; t < 24; t += 8) {
        const int mat = t >> 3;
        const int n0  = (t & 7) * 16;
        const float* __restrict__ B = (mat == 0) ? Bq : ((mat == 1) ? Bk : Bv);
        v8f c = {};
        #pragma unroll 8
        for (int kk = 0; kk < 128; kk += 4) {
            const int krow = kk + 2 * half;
            v2f a, b;
            a.x = As[l16 * 132 + krow];
            a.y = As[l16 * 132 + krow + 1];
            b.x = B[(size_t)krow       * 128 + n0 + l16];
            b.y = B[(size_t)(krow + 1) * 128 + n0 + l16];
            c = __builtin_amdgcn_wmma_f32_16x16x4_f32(
                    false, a, false, b, (short)0, c, false, false);
        }
        float* __restrict__ D = (mat == 0) ? Q : ((mat == 1) ? K : V);
        #pragma unroll
        for (int i = 0; i < 8; i++) {
            const int row = i + 8 * half;
            D[(size_t)(m0 + row) * 128 + n0 + l16] = c[i];
        }
    }
}

// =====================================================================
// Per-layer scratch init.
// =====================================================================
__global__ __launch_bounds__(256) void init_layer(unsigned* __restrict__ mkey,
                                                  float* __restrict__ denom,
                                                  float* __restrict__ outc)
{
    const int i = blockIdx.x * 256 + threadIdx.x;
    if (i < N_NODES * HEADS) { mkey[i] = 0u; denom[i] = 0.0f; }
    if (i < N_NODES * HC)    outc[i] = 0.0f;
}

__global__ __launch_bounds__(256) void init_globals(float* __restrict__ g123,
                                                    float* __restrict__ cnt)
{
    const int i = blockIdx.x * 256 + threadIdx.x;
    if (i < 3 * N_GRAPHS * DIM_H) g123[i] = 0.0f;
    if (i < N_GRAPHS)             cnt[i]  = 0.0f;
}

// =====================================================================
// Edge logits: one wave32 per (edge, head); lane == channel (C==32).
// edge_attr row read as 4 x b128; e_feat recomputed (16 MACs).
// Segment-max via monotonic-key unsigned atomicMax.
// =====================================================================
__global__ __launch_bounds__(256) void edge_logits(
    const float* __restrict__ Q, const float* __restrict__ Kt,
    const float* __restrict__ EA, const float* __restrict__ We,
    const int* __restrict__ src, const int* __restrict__ dst,
    float* __restrict__ logits, unsigned* __restrict__ mkey)
{
    const int wave = threadIdx.x >> 5;
    const int lane = threadIdx.x & 31;
    const long task = (long)blockIdx.x * 8 + wave;           // < E*HEADS
    const int e = (int)(task >> 2);
    const int h = (int)(task & 3);
    const int s = src[e], d = dst[e];

    const float4* __restrict__ ea4 = (const float4*)(EA + (size_t)e * EDGE_DIM);
    const float4 w0 = ea4[0], w1 = ea4[1], w2 = ea4[2], w3 = ea4[3];
    const float eav[16] = {w0.x, w0.y, w0.z, w0.w, w1.x, w1.y, w1.z, w1.w,
                           w2.x, w2.y, w2.z, w2.w, w3.x, w3.y, w3.z, w3.w};
    float ef = 0.0f;
    #pragma unroll
    for (int t = 0; t < EDGE_DIM; t++)
        ef = fmaf(eav[t], We[t * HC + h * DIM_H + lane], ef);

    const float qv = Q [(size_t)d * HC + h * DIM_H + lane];
    const float kv = Kt[(size_t)s * HC + h * DIM_H + lane];
    float part = qv * (kv + ef);
    #pragma unroll
    for (int off = 16; off > 0; off >>= 1)
        part += __shfl_xor(part, off, 32);
    part *= 0.17677669529663687f;                            // 1/sqrt(32)

    if (lane == 0) {
        logits[(size_t)e * HEADS + h] = part;
        atomicMax(&mkey[(size_t)d * HEADS + h], fkey(part));
    }
}

// Convert max keys back to floats; empty segments (key==0 -> -NaN) -> 0.
__global__ __launch_bounds__(256) void m_fix(unsigned* __restrict__ mkey)
{
    const int i = blockIdx.x * 256 + threadIdx.x;
    if (i >= N_NODES * HEADS) return;
    const float f = funkey(mkey[i]);
    ((float*)mkey)[i] = isfinite(f) ? f : 0.0f;
}

// p = exp(logit - m[dst]) in place; denom[dst] += p.
__global__ __launch_bounds__(256) void p_denom(
    float* __restrict__ logits, const float* __restrict__ m,
    float* __restrict__ denom, const int* __restrict__ dst)
{
    const int i = blockIdx.x * 256 + threadIdx.x;            // < E*HEADS
    if (i >= N_EDGES * HEADS) return;
    const int e = i >> 2, h = i & 3;
    const int d = dst[e];
    const float p = __expf(logits[i] - m[(size_t)d * HEADS + h]);
    logits[i] = p;
    atomicAdd(&denom[(size_t)d * HEADS + h], p);
}

// Aggregation: 2 edges/block, 128 threads/edge (thread == output channel).
__global__ __launch_bounds__(256) void aggregate(
    const float* __restrict__ p, const float* __restrict__ denom,
    const float* __restrict__ V, const float* __restrict__ EA,
    const float* __restrict__ We,
    const int* __restrict__ src, const int* __restrict__ dst,
    float* __restrict__ outc)
{
    const long e = (long)blockIdx.x * 2 + (threadIdx.x >> 7);
    if (e >= N_EDGES) return;
    const int ch = threadIdx.x & 127;
    const int h  = ch >> 5;
    const int s = src[e], d = dst[e];

    const float4* __restrict__ ea4 = (const float4*)(EA + (size_t)e * EDGE_DIM);
    const float4 w0 = ea4[0], w1 = ea4[1], w2 = ea4[2], w3 = ea4[3];
    const float eav[16] = {w0.x, w0.y, w0.z, w0.w, w1.x, w1.y, w1.z, w1.w,
                           w2.x, w2.y, w2.z, w2.w, w3.x, w3.y, w3.z, w3.w};
    float ef = 0.0f;
    #pragma unroll
    for (int t = 0; t < EDGE_DIM; t++)
        ef = fmaf(eav[t], We[t * HC + ch], ef);

    const float alpha = p[(size_t)e * HEADS + h] /
                        (denom[(size_t)d * HEADS + h] + 1e-16f);
    atomicAdd(&outc[(size_t)d * HC + ch],
              alpha * (V[(size_t)s * HC + ch] + ef));
}

__global__ __launch_bounds__(256) void relu_copy(const float* __restrict__ a,
                                                 float* __restrict__ b, int n)
{
    const int i = blockIdx.x * 256 + threadIdx.x;
    if (i < n) b[i] = fmaxf(a[i], 0.0f);
}

// Layer 3: h3 = relu(mean over heads).
__global__ __launch_bounds__(256) void meanheads_relu(const float* __restrict__ outc,
                                                      float* __restrict__ h3)
{
    const int i = blockIdx.x * 256 + threadIdx.x;            // < N*DIM_H
    if (i >= N_NODES * DIM_H) return;
    const int n = i >> 5, c = i & 31;
    const size_t b = (size_t)n * HC + c;
    const float v = 0.25f * (outc[b] + outc[b + 32] + outc[b + 64] + outc[b + 96]);
    h3[i] = fmaxf(v, 0.0f);
}

// Graph pooling (sum); hc==128 -> mean over heads first, hc==32 -> direct.
__global__ __launch_bounds__(256) void pool_nodes(const float* __restrict__ h,
                                                  const int* __restrict__ batch,
                                                  float* __restrict__ g, int hc)
{
    const int i = blockIdx.x * 256 + threadIdx.x;            // < N*DIM_H
    if (i >= N_NODES * DIM_H) return;
    const int n = i >> 5, c = i & 31;
    float v;
    if (hc == HC) {
        const size_t b = (size_t)n * HC + c;
        v = 0.25f * (h[b] + h[b + 32] + h[b + 64] + h[b + 96]);
    } else {
        v = h[(size_t)n * DIM_H + c];
    }
    atomicAdd(&g[(size_t)batch[n] * DIM_H + c], v);
}

__global__ __launch_bounds__(256) void count_nodes(const int* __restrict__ batch,
                                                   float* __restrict__ cnt)
{
    const int i = blockIdx.x * 256 + threadIdx.x;
    if (i < N_NODES) atomicAdd(&cnt[batch[i]], 1.0f);
}

// =====================================================================
// Head MLP + log_softmax over 64 graphs (tiny; one block).
// =====================================================================
__global__ __launch_bounds__(256) void mlp_head(
    const float* __restrict__ g123, const float* __restrict__ cnt,
    const float* __restrict__ w1, const float* __restrict__ b1,
    const float* __restrict__ w2, const float* __restrict__ b2,
    float* __restrict__ out)
{
    __shared__ float G[N_GRAPHS * 96];
    __shared__ float H[N_GRAPHS * 96];
    __shared__ float O[N_GRAPHS * FINAL_DIM];
    const int tid = threadIdx.x;

    for (int i = tid; i < N_GRAPHS * 96; i += 256) {
        const int r = i / 96, c = i % 96;
        const int part = c >> 5, cc = c & 31;
        float cn = cnt[r]; cn = (cn < 1.0f) ? 1.0f : cn;
        G[i] = g123[(size_t)part * N_GRAPHS * DIM_H + r * DIM_H + cc] / cn;
    }
    __syncthreads();
    for (int i = tid; i < N_GRAPHS * 96; i += 256) {
        const int r = i / 96, c = i % 96;
        float acc = b1[c];
        #pragma unroll 8
        for (int k = 0; k < 96; k++) acc = fmaf(G[r * 96 + k], w1[k * 96 + c], acc);
        H[i] = fmaxf(acc, 0.0f);
    }
    __syncthreads();
    for (int i = tid; i < N_GRAPHS * FINAL_DIM; i += 256) {
        const int r = i / FINAL_DIM, c = i % FINAL_DIM;
        float acc = b2[c];
        #pragma unroll 8
        for (int k = 0; k < 96; k++) acc = fmaf(H[r * 96 + k], w2[k * FINAL_DIM + c], acc);
        O[i] = acc;
    }
    __syncthreads();
    if (tid < N_GRAPHS) {
        float mx = -INFINITY;
        #pragma unroll
        for (int c = 0; c < FINAL_DIM; c++) mx = fmaxf(mx, O[tid * FINAL_DIM + c]);
        float s = 0.0f;
        #pragma unroll
        for (int c = 0; c < FINAL_DIM; c++) s += __expf(O[tid * FINAL_DIM + c] - mx);
        const float lse = mx + __logf(s);
        #pragma unroll
        for (int c = 0; c < FINAL_DIM; c++)
            out[tid * FINAL_DIM + c] = O[tid * FINAL_DIM + c] - lse;
    }
}

// =====================================================================
extern "C" void kernel_launch(void* const* d_in, const int* in_sizes, int n_in,
                              void* d_out, int out_size, void* d_ws, size_t ws_size,
                              hipStream_t stream)
{
    const float* x     = (const float*)d_in[0];
    const float* ea    = (const float*)d_in[1];
    const int*   eidx  = (const int*)  d_in[2];
    const int*   batch = (const int*)  d_in[3];
    const float* WQ[3] = {(const float*)d_in[4],  (const float*)d_in[8],  (const float*)d_in[12]};
    const float* WK[3] = {(const float*)d_in[5],  (const float*)d_in[9],  (const float*)d_in[13]};
    const float* WV[3] = {(const float*)d_in[6],  (const float*)d_in[10], (const float*)d_in[14]};
    const float* WE[3] = {(const float*)d_in[7],  (const float*)d_in[11], (const float*)d_in[15]};
    const float* lin1w = (const float*)d_in[16];
    const float* lin1b = (const float*)d_in[17];
    const float* lin2w = (const float*)d_in[18];
    const float* lin2b = (const float*)d_in[19];
    const int* src = eidx;
    const int* dst = eidx + N_EDGES;

    // workspace carve-up (floats)
    const size_t NHC = (size_t)N_NODES * HC;       // 2,560,000
    const size_t EH  = (size_t)N_EDGES * HEADS;    // 1,280,000
    const size_t NH  = (size_t)N_NODES * HEADS;    //    80,000
    float* ws     = (float*)d_ws;
    float* q      = ws;
    float* k      = q + NHC;
    float* v      = k + NHC;
    float* logits = v + NHC;                       // EH (reused in place for p)
    float* m      = logits + EH;                   // NH (uint keys, then floats)
    float* denom  = m + NH;                        // NH
    float* outc   = denom + NH;                    // NHC
    float* h1     = outc + NHC;                    // NHC
    float* h2     = h1 + NHC;                      // NHC
    float* h3     = h2 + NHC;                      // N*DIM_H
    float* g123   = h3 + (size_t)N_NODES * DIM_H;  // 3*64*32
    float* cnt    = g123 + 3 * N_GRAPHS * DIM_H;   // 64

    const dim3 B256(256);
    const dim3 G_gemm(N_NODES / 16);               // 1250
    const dim3 G_nhc((N_NODES * HC) / 256);        // 10000
    const dim3 G_ewave((N_EDGES * HEADS) / 8);     // 160000 (wave per edge-head)
    const dim3 G_eh((N_EDGES * HEADS) / 256);      // 5000
    const dim3 G_agg(N_EDGES / 2);                 // 160000
    const dim3 G_nh((N_NODES * HEADS + 255) / 256);
    const dim3 G_nd((N_NODES * DIM_H) / 256);      // 2500
    const dim3 G_n((N_NODES + 255) / 256);

    init_globals<<<dim3((3 * N_GRAPHS * DIM_H + 255) / 256), B256, 0, stream>>>(g123, cnt);
    count_nodes<<<G_n, B256, 0, stream>>>(batch, cnt);

    const float* layer_in[3] = {x, h1, h2};
    float*       layer_h [3] = {h1, h2, h3};

    for (int L = 0; L < 3; L++) {
        qkv_wmma<<<G_gemm, B256, 0, stream>>>(layer_in[L], WQ[L], WK[L], WV[L], q, k, v);
        init_layer<<<G_nhc, B256, 0, stream>>>((unsigned*)m, denom, outc);
        edge_logits<<<G_ewave, B256, 0, stream>>>(q, k, ea, WE[L], src, dst,
                                                  logits, (unsigned*)m);
        m_fix<<<G_nh, B256, 0, stream>>>((unsigned*)m);
        p_denom<<<G_eh, B256, 0, stream>>>(logits, m, denom, dst);
        aggregate<<<G_agg, B256, 0, stream>>>(logits, denom, v, ea, WE[L], src, dst, outc);
        if (L < 2) {
            relu_copy<<<G_nhc, B256, 0, stream>>>(outc, layer_h[L], N_NODES * HC);
            pool_nodes<<<G_nd, B256, 0, stream>>>(layer_h[L], batch,
                                                  g123 + (size_t)L * N_GRAPHS * DIM_H, HC);
        } else {
            meanheads_relu<<<G_nd, B256, 0, stream>>>(outc, h3);
            pool_nodes<<<G_nd, B256, 0, stream>>>(h3, batch,
                                                  g123 + 2 * (size_t)N_GRAPHS * DIM_H, DIM_H);
        }
    }

    mlp_head<<<dim3(1), B256, 0, stream>>>(g123, cnt, lin1w, lin1b, lin2w, lin2b,
                                           (float*)d_out);
}